// XY2UV_38946763440612
// MI455X (gfx1250) — compile-verified
//
#include <hip/hip_runtime.h>

#define BQ  4
#define CQ  3
#define HQ  512
#define WQ  512
#define VQ  12000
#define NFQ 24000
#define UHQ 1024
#define UWQ 1024

typedef __attribute__((ext_vector_type(2))) float v2f;
typedef __attribute__((ext_vector_type(8))) float v8f;

// ---------------------------------------------------------------------------
// Kernel 1: mesh_cam = R*mesh + t  via V_WMMA_F32_16X16X4_F32 (homogeneous),
// then pinhole projection to mesh_img (B,V,2).
// Each wave handles 16 vertices of one batch. EXEC is all-ones through the
// WMMA (wave-uniform early-out only).
// ---------------------------------------------------------------------------
__global__ __launch_bounds__(128) void project_kernel(
    const float* __restrict__ mesh,    // (B,V,3)
    const float* __restrict__ focal,   // (B,2)
    const float* __restrict__ princpt, // (B,2)
    const float* __restrict__ Rm,      // (B,3,3)
    const float* __restrict__ tv,      // (B,3)
    float* __restrict__ mesh_img)      // (B,V,2)
{
  const int tilesPerB = (VQ + 15) / 16;
  const int wave  = blockIdx.x * (blockDim.x >> 5) + (threadIdx.x >> 5);
  const int lane  = threadIdx.x & 31;
  const int wslot = threadIdx.x >> 5;
  const int kgrp  = lane >> 4;   // 0: lanes 0-15, 1: lanes 16-31
  const int n     = lane & 15;

  __shared__ float cam[4][16][4];

  if (wave < BQ * tilesPerB) {
    const int b     = wave / tilesPerB;
    const int vbase = (wave % tilesPerB) * 16;

    // A (16x4): row m = vertex vbase+m, cols = [x, y, z, 1]
    // lanes 0-15: VGPR0=K0, VGPR1=K1 ; lanes 16-31: VGPR0=K2, VGPR1=K3
    int vm = vbase + n; if (vm >= VQ) vm = VQ - 1;
    const float* mp = mesh + ((size_t)b * VQ + vm) * 3;
    v2f a;
    if (kgrp == 0) { a.x = mp[0]; a.y = mp[1]; }
    else           { a.x = mp[2]; a.y = 1.0f;  }

    // B (4x16): B[k][n] = (k<3) ? R[n][k] : t[n]; columns n>=3 are zero.
    // lanes 0-15: VGPR0=K0 row, VGPR1=K1 row ; lanes 16-31: K2, K3.
    const float* Rb = Rm + b * 9;
    const float* tb = tv + b * 3;
    v2f bm; bm.x = 0.0f; bm.y = 0.0f;
    if (n < 3) {
      if (kgrp == 0) { bm.x = Rb[n * 3 + 0]; bm.y = Rb[n * 3 + 1]; }
      else           { bm.x = Rb[n * 3 + 2]; bm.y = tb[n];         }
    }

    v8f c = {};
    v8f d = __builtin_amdgcn_wmma_f32_16x16x4_f32(
        /*neg_a=*/false, a, /*neg_b=*/false, bm,
        /*c_mod=*/(short)0, c, /*reuse_a=*/false, /*reuse_b=*/false);

    // D layout: VGPR r -> row r (lanes 0-15) / row 8+r (lanes 16-31), col = n.
    // Stage columns 0..2 (X,Y,Z) into LDS to re-shard columns -> lanes.
    if (n < 3) {
#pragma unroll
      for (int r = 0; r < 8; ++r) cam[wslot][kgrp * 8 + r][n] = d[r];
    }
  }
  __syncthreads();

  if (wave < BQ * tilesPerB && lane < 16) {
    const int b     = wave / tilesPerB;
    const int vbase = (wave % tilesPerB) * 16;
    const int v     = vbase + lane;
    if (v < VQ) {
      float X = cam[wslot][lane][0];
      float Y = cam[wslot][lane][1];
      float Z = cam[wslot][lane][2];
      float fx = focal[b * 2 + 0],   fy = focal[b * 2 + 1];
      float cx = princpt[b * 2 + 0], cy = princpt[b * 2 + 1];
      float xi = X / Z * fx + cx;
      float yi = Y / Z * fy + cy;
      float* o = mesh_img + ((size_t)b * VQ + v) * 2;
      o[0] = xi;
      o[1] = yi;
    }
  }
}

// ---------------------------------------------------------------------------
// Kernel 2a/2b: visibility mask (B,NF). Reference quirk: pix==-1 -> local
// wraps to NF-1. Races write the same value (1), so plain stores are exact.
// ---------------------------------------------------------------------------
__global__ void clear_valid_kernel(int* __restrict__ valid) {
  int i = blockIdx.x * blockDim.x + threadIdx.x;
  if (i < BQ * NFQ) valid[i] = 0;
}

__global__ void scatter_valid_kernel(const int* __restrict__ p2f_xy,
                                     int* __restrict__ valid) {
  int i = blockIdx.x * blockDim.x + threadIdx.x;
  if (i >= BQ * HQ * WQ) return;
  int b = i / (HQ * WQ);
  int p = p2f_xy[i];
  int local = (p == -1) ? (NFQ - 1) : (p - NFQ * b);
  if (local < 0) local += NFQ;  // matches reference wrap semantics
  valid[b * NFQ + local] = 1;
}

// ---------------------------------------------------------------------------
// Kernel 3: per (b, uv-pixel): gather face -> vertices -> projected pts,
// barycentric combine, replicate reference grid normalize/denormalize,
// bilinear sample (zero-pad OOB), masking collapses to -1 when not visible.
// ---------------------------------------------------------------------------
__global__ __launch_bounds__(256) void uv_main_kernel(
    const float* __restrict__ img,      // (B,C,H,W)
    const int*   __restrict__ face,     // (NF,3)
    const float* __restrict__ mesh_img, // (B,V,2)
    const int*   __restrict__ valid,    // (B,NF)
    const int*   __restrict__ puv_map,  // (UH,UW)
    const float* __restrict__ bary,     // (UH,UW,3)
    float*       __restrict__ out)      // (B,C,UH,UW)
{
  int idx = blockIdx.x * blockDim.x + threadIdx.x;
  if (idx >= BQ * UHQ * UWQ) return;
  int xq = idx & (UWQ - 1);
  int yq = (idx >> 10) & (UHQ - 1);
  int b  = idx >> 20;

  int praw = puv_map[yq * UWQ + xq];
  int puv  = (praw < 0) ? praw + NFQ : praw;
  float o0 = -1.0f, o1 = -1.0f, o2 = -1.0f;

  bool visible = (praw != -1) && (valid[b * NFQ + puv] != 0);
  if (visible) {
    const int* f = face + (size_t)puv * 3;
    int v0 = f[0], v1 = f[1], v2 = f[2];
    const float* mi = mesh_img + (size_t)b * VQ * 2;
    float g0x = mi[2 * v0], g0y = mi[2 * v0 + 1];
    float g1x = mi[2 * v1], g1y = mi[2 * v1 + 1];
    float g2x = mi[2 * v2], g2y = mi[2 * v2 + 1];
    const float* bw = bary + ((size_t)yq * UWQ + xq) * 3;
    float w0 = bw[0], w1 = bw[1], w2 = bw[2];
    float u  = w0 * g0x + w1 * g1x + w2 * g2x;
    float vv = w0 * g0y + w1 * g1y + w2 * g2y;

    // replicate reference: normalize to grid then denormalize
    float gx = u  / (float)(WQ - 1) * 2.0f - 1.0f;
    float gy = vv / (float)(HQ - 1) * 2.0f - 1.0f;
    float ix = (gx + 1.0f) * 0.5f * (float)(WQ - 1);
    float iy = (gy + 1.0f) * 0.5f * (float)(HQ - 1);

    float ix0 = floorf(ix), iy0 = floorf(iy);
    float wx1 = ix - ix0, wx0 = 1.0f - wx1;
    float wy1 = iy - iy0, wy0 = 1.0f - wy1;

    float a0 = 0.0f, a1 = 0.0f, a2 = 0.0f;
    const float* ib = img + (size_t)b * CQ * HQ * WQ;
    auto tap = [&](float yf, float xf, float w) {
      bool ok = (xf >= 0.0f) && (xf <= (float)(WQ - 1)) &&
                (yf >= 0.0f) && (yf <= (float)(HQ - 1));
      float xc = fminf(fmaxf(xf, 0.0f), (float)(WQ - 1));
      float yc = fminf(fmaxf(yf, 0.0f), (float)(HQ - 1));
      int xi = (int)xc, yi = (int)yc;
      float ww = ok ? w : 0.0f;
      size_t o = (size_t)yi * WQ + xi;
      a0 += ib[o] * ww;
      a1 += ib[(size_t)HQ * WQ + o] * ww;
      a2 += ib[2 * (size_t)HQ * WQ + o] * ww;
    };
    tap(iy0,        ix0,        wy0 * wx0);
    tap(iy0,        ix0 + 1.0f, wy0 * wx1);
    tap(iy0 + 1.0f, ix0,        wy1 * wx0);
    tap(iy0 + 1.0f, ix0 + 1.0f, wy1 * wx1);
    o0 = a0; o1 = a1; o2 = a2;
  }

  size_t ob = (size_t)b * CQ * UHQ * UWQ + (size_t)yq * UWQ + xq;
  out[ob]                           = o0;
  out[ob + (size_t)UHQ * UWQ]       = o1;
  out[ob + 2 * (size_t)UHQ * UWQ]   = o2;
}

// ---------------------------------------------------------------------------
extern "C" void kernel_launch(void* const* d_in, const int* in_sizes, int n_in,
                              void* d_out, int out_size, void* d_ws, size_t ws_size,
                              hipStream_t stream) {
  const float* img     = (const float*)d_in[0];
  const float* mesh    = (const float*)d_in[1];
  const int*   face    = (const int*)  d_in[2];
  const float* focal   = (const float*)d_in[3];
  const float* princpt = (const float*)d_in[4];
  const float* Rm      = (const float*)d_in[5];
  const float* tv      = (const float*)d_in[6];
  const int*   p2f_xy  = (const int*)  d_in[7];
  const int*   p2f_uv  = (const int*)  d_in[8];
  const float* bary    = (const float*)d_in[9];
  float* out = (float*)d_out;

  // scratch layout: mesh_img (B*V*2 f32) | valid (B*NF i32)
  char*  ws       = (char*)d_ws;
  float* mesh_img = (float*)ws;
  int*   valid    = (int*)(ws + sizeof(float) * (size_t)BQ * VQ * 2);

  const int tilesPerB = (VQ + 15) / 16;
  const int waves     = BQ * tilesPerB;          // 3000 waves, 4 per block
  project_kernel<<<(waves + 3) / 4, 128, 0, stream>>>(
      mesh, focal, princpt, Rm, tv, mesh_img);

  clear_valid_kernel<<<(BQ * NFQ + 255) / 256, 256, 0, stream>>>(valid);
  scatter_valid_kernel<<<(BQ * HQ * WQ + 255) / 256, 256, 0, stream>>>(
      p2f_xy, valid);

  uv_main_kernel<<<(BQ * UHQ * UWQ + 255) / 256, 256, 0, stream>>>(
      img, face, mesh_img, valid, p2f_uv, bary, out);
}